// HashGrid_38379827757290
// MI455X (gfx1250) — compile-verified
//
#include <hip/hip_runtime.h>
#include <hip/hip_bf16.h>

typedef float v4f __attribute__((ext_vector_type(4)));
typedef int   v4i __attribute__((ext_vector_type(4)));
typedef __attribute__((address_space(1))) int gint;   // global (AS1)
typedef __attribute__((address_space(3))) int lint;   // LDS (AS3)

#ifndef __has_builtin
#define __has_builtin(x) 0
#endif

#if __has_builtin(__builtin_amdgcn_global_load_async_to_lds_b32)
#define HAVE_ASYNC_LDS 1
#else
#define HAVE_ASYNC_LDS 0
#endif

#if __has_builtin(__builtin_amdgcn_s_wait_asynccnt)
#define HAVE_WAIT_ASYNCCNT 1
#else
#define HAVE_WAIT_ASYNCCNT 0
#endif

#define LLV 16
#define FFEAT 28
#define PTS_PER_BLOCK 32
#define THREADS 256

// Raw metadata staged by async DMA: [0..16] offsets, [17..64] resolution_list,
// [65..80] side_length bits. Then 16 threads pack per-level records:
// s_lvl[l*8 + {0:side,1:r0,2:r1,3:r2,4:base,5:mask,6:sz,7:direct}]
__global__ __launch_bounds__(THREADS) void hashgrid_encode_kernel(
    const float* __restrict__ inputs,
    const float* __restrict__ memory,
    const int*   __restrict__ offsets,
    const int*   __restrict__ res_list,
    const float* __restrict__ side_list,
    float* __restrict__ out,
    int n_points)
{
    __shared__ int   s_meta[17 + 48 + 16];
    __shared__ __attribute__((aligned(16))) int s_lvl[LLV * 8];
    __shared__ float s_pts[PTS_PER_BLOCK * 3];

    const int tid = threadIdx.x;

#if HAVE_ASYNC_LDS
    // Async DMA (gfx1250 GLOBAL_LOAD_ASYNC_TO_LDS_B32) of per-level metadata
    // and this block's 32 points into LDS; tracked with ASYNCcnt.
    if (tid < 81) {
        const int* g = (tid < 17) ? (offsets + tid)
                     : (tid < 65) ? (res_list + (tid - 17))
                     : ((const int*)side_list + (tid - 65));
        __builtin_amdgcn_global_load_async_to_lds_b32(
            (gint*)g, (lint*)&s_meta[tid], 0, 0);
    } else if (tid >= 96 && tid < 96 + PTS_PER_BLOCK * 3) {
        const int j  = tid - 96;
        const int gi = blockIdx.x * (PTS_PER_BLOCK * 3) + j;
        if (gi < n_points * 3) {
            __builtin_amdgcn_global_load_async_to_lds_b32(
                (gint*)(inputs + gi), (lint*)&s_pts[j], 0, 0);
        }
    }
#if HAVE_WAIT_ASYNCCNT
    __builtin_amdgcn_s_wait_asynccnt(0);
#else
    asm volatile("s_wait_asynccnt 0" ::: "memory");
#endif
#else
    if (tid < 81) {
        s_meta[tid] = (tid < 17) ? offsets[tid]
                    : (tid < 65) ? res_list[tid - 17]
                    : ((const int*)side_list)[tid - 65];
    } else if (tid >= 96 && tid < 96 + PTS_PER_BLOCK * 3) {
        const int j  = tid - 96;
        const int gi = blockIdx.x * (PTS_PER_BLOCK * 3) + j;
        if (gi < n_points * 3) s_pts[j] = inputs[gi];
    }
#endif
    __syncthreads();

    // Pack per-level constants once per block (16 threads).
    if (tid < LLV) {
        const int base = s_meta[tid];
        const int size = s_meta[tid + 1] - base;
        const int r0 = s_meta[17 + tid * 3 + 0];
        const int r1 = s_meta[17 + tid * 3 + 1];
        const int r2 = s_meta[17 + tid * 3 + 2];
        // Reference: prod(res.f32) <= size.f32 selects direct indexing.
        const int direct = ((((float)r0 * (float)r1) * (float)r2) <= (float)size) ? 1 : 0;
        s_lvl[tid * 8 + 0] = s_meta[65 + tid];       // side (float bits)
        s_lvl[tid * 8 + 1] = r0;
        s_lvl[tid * 8 + 2] = r1;
        s_lvl[tid * 8 + 3] = r2;
        s_lvl[tid * 8 + 4] = base;
        // When !direct, size == min(T, prod) == T == 524288 (power of two),
        // so hashed % size is exactly & (size-1).
        s_lvl[tid * 8 + 5] = size - 1;               // mask
        s_lvl[tid * 8 + 6] = r0 * r1;                // z-stride for direct
        s_lvl[tid * 8 + 7] = direct;
    }
    __syncthreads();

    const int group = tid >> 3;   // 32 point-groups per block
    const int lane  = tid & 7;    // 8 lanes per point; lanes 0..6 carry float4 of features
    const int n     = blockIdx.x * PTS_PER_BLOCK + group;
    if (n >= n_points) return;

    // BOUNDS mins hardcoded to match the reference constants.
    const float px = s_pts[group * 3 + 0] - (-3.0f);
    const float py = s_pts[group * 3 + 1] - (-4.0f);
    const float pz = s_pts[group * 3 + 2] - (-2.0f);

    const int flane = (lane < 7) ? lane : 0;  // lane 7 re-reads lane 0's slot (discarded)
    const int foff  = flane * 4;
    float* outp = out + (size_t)n * (LLV * FFEAT);

    for (int l = 0; l < LLV; ++l) {
        const v4i m0 = *(const v4i*)&s_lvl[l * 8];
        const v4i m1 = *(const v4i*)&s_lvl[l * 8 + 4];
        const float side = __int_as_float(m0.x);
        const int r0 = m0.y, r1 = m0.z, r2 = m0.w;
        const int base = m1.x;
        const unsigned mask = (unsigned)m1.y;
        const unsigned sz   = (unsigned)m1.z;
        const bool direct   = m1.w != 0;
        const unsigned sy   = (unsigned)r0;

        const float qx = px / side, qy = py / side, qz = pz / side;
        const float gx = floorf(qx), gy = floorf(qy), gz = floorf(qz);
        const float fx = qx - gx, fy = qy - gy, fz = qz - gz;
        const int ix = (int)gx, iy = (int)gy, iz = (int)gz;

        unsigned idxc[8];
        float    wc[8];
#pragma unroll
        for (int c = 0; c < 8; ++c) {
            const int dx = c & 1, dy = (c >> 1) & 1, dz = (c >> 2) & 1;
            int cx = ix + dx; cx = cx < 0 ? 0 : (cx > r0 - 1 ? r0 - 1 : cx);
            int cy = iy + dy; cy = cy < 0 ? 0 : (cy > r1 - 1 ? r1 - 1 : cy);
            int cz = iz + dz; cz = cz < 0 ? 0 : (cz > r2 - 1 ? r2 - 1 : cz);
            const unsigned ux = (unsigned)cx, uy = (unsigned)cy, uz = (unsigned)cz;
            unsigned id;
            if (direct) {
                id = ux + uy * sy + uz * sz;
            } else {
                const unsigned h = ux * 1u ^ uy * 2654435761u ^ uz * 805459861u;
                id = h & mask;
            }
            idxc[c] = (unsigned)base + id;
            wc[c] = (dx ? fx : 1.0f - fx) * (dy ? fy : 1.0f - fy) * (dz ? fz : 1.0f - fz);
        }

        // Issue all 8 coalesced 112B gathers before consuming (max loads in flight).
        v4f vals[8];
#pragma unroll
        for (int c = 0; c < 8; ++c) {
            vals[c] = *(const v4f*)(memory + (size_t)idxc[c] * FFEAT + foff);
        }

        v4f acc = {0.0f, 0.0f, 0.0f, 0.0f};
#pragma unroll
        for (int c = 0; c < 8; ++c) acc += vals[c] * wc[c];

        if (lane < 7) {
            // Streaming output (470MB): non-temporal so hash tables stay in L2.
            __builtin_nontemporal_store(acc, (v4f*)(outp + l * FFEAT + foff));
        }
    }
}

extern "C" void kernel_launch(void* const* d_in, const int* in_sizes, int n_in,
                              void* d_out, int out_size, void* d_ws, size_t ws_size,
                              hipStream_t stream) {
    const float* inputs    = (const float*)d_in[0];
    const float* memory    = (const float*)d_in[1];
    const int*   offsets   = (const int*)d_in[2];
    const int*   res_list  = (const int*)d_in[3];
    const float* side_list = (const float*)d_in[4];
    float* out = (float*)d_out;

    const int n_points = in_sizes[0] / 3;
    const int blocks = (n_points + PTS_PER_BLOCK - 1) / PTS_PER_BLOCK;
    hashgrid_encode_kernel<<<blocks, THREADS, 0, stream>>>(
        inputs, memory, offsets, res_list, side_list, out, n_points);
}